// HybridFFT_33071248180104
// MI455X (gfx1250) — compile-verified
//
#include <hip/hip_runtime.h>

// CDNA5 / gfx1250 Walsh-Hadamard transform via WMMA.
// y = (H16 (x) H16 (x) H16) x per 4096-float row; 8192 rows.
// One wave32 processes one row: 3 mode contractions x 16 tiles x 4
// v_wmma_f32_16x16x4_f32 = 192 WMMA per row, full f32 precision.
// Memory-bound: 256 MB HBM traffic -> ~11 us floor at 23.3 TB/s.

typedef __attribute__((ext_vector_type(2))) float v2f;
typedef __attribute__((ext_vector_type(8))) float v8f;

#define WAVES_PER_BLOCK 4
// LDS tensor layout: addr(i,k,j) = i*SI + k*SK + j  (floats)
// SI=257 (odd) and SK=17 (coprime with 64 banks) break the 8-way
// bank conflicts that strides 256/16 would cause on the m*16 patterns.
#define SI 257
#define SK 17
#define LDS_PER_WAVE (15 * SI + 15 * SK + 15 + 1)   // 4126 floats

__global__ __launch_bounds__(WAVES_PER_BLOCK * 32) void
wht4096_wmma_kernel(const float* __restrict__ x, float* __restrict__ y, int rows) {
  __shared__ float smem[WAVES_PER_BLOCK * LDS_PER_WAVE];

  const int lane = threadIdx.x & 31;
  const int wave = threadIdx.x >> 5;
  const int m = lane & 15;   // M / N index within a 16-wide tile
  const int g = lane >> 4;   // half-wave select
  float* __restrict__ lds = smem + wave * LDS_PER_WAVE;

  const int row = blockIdx.x * WAVES_PER_BLOCK + wave;
  if (row >= rows) return;   // uniform per wave -> EXEC stays all-1s for WMMA

  const float* __restrict__ xin = x + (size_t)row * 4096;
  float* __restrict__       yo  = y + (size_t)row * 4096;

  // A operand = H16 K-slices, built in-register.
  // ISA 16x4 f32 A layout: lane L, vgpr v holds (M = L&15, K = 2*(L>>4)+v);
  // slice s covers global K = 4s..4s+3, so element = H[m][4s+2g+v],
  // H[a][b] = (-1)^popcount(a&b).
  v2f A[4];
#pragma unroll
  for (int s = 0; s < 4; ++s) {
    const int k0 = 4 * s + 2 * g;
    A[s].x = (__builtin_popcount(m & k0) & 1) ? -1.0f : 1.0f;
    A[s].y = (__builtin_popcount(m & (k0 + 1)) & 1) ? -1.0f : 1.0f;
  }

  // ---------------- Stage 1: contract k (global -> WMMA -> LDS) -------------
  // Input is k-fastest, so B (4x16: K=2g+v, N=L&15) is a direct b64 load:
  // B[k][n=j] = x[i=blk][j][k].  D[m=k'][n=j] = sum_k H[k'][k] x[blk][j][k].
#pragma unroll
  for (int blk = 0; blk < 16; ++blk) {
    v8f acc = {};
#pragma unroll
    for (int s = 0; s < 4; ++s) {
      const v2f b = __builtin_nontemporal_load(
          (const v2f*)(xin + blk * 256 + m * 16 + 4 * s + 2 * g));
      acc = __builtin_amdgcn_wmma_f32_16x16x4_f32(
          false, A[s], false, b, (short)0, acc, false, false);
    }
    // store y1[i=blk][j][k'] at addr(i,k',j); D: vgpr v, lane L -> (k'=v+8g, j=L&15)
#pragma unroll
    for (int v = 0; v < 8; ++v)
      lds[blk * SI + (v + 8 * g) * SK + m] = acc[v];
  }

  // ---------------- Stage 2: contract j (LDS in-place) ----------------------
  // Tile blk = k'. B[j][n=i] = y1[i=L&15][j=4s+2g+v][blk]  (j is fastest).
  // Reads and writes of a tile hit the same disjoint address set; the
  // acc dependence orders all reads before any write -> in-place is safe.
#pragma unroll
  for (int blk = 0; blk < 16; ++blk) {
    v8f acc = {};
#pragma unroll
    for (int s = 0; s < 4; ++s) {
      v2f b;
      b.x = lds[m * SI + blk * SK + (4 * s + 2 * g)];
      b.y = lds[m * SI + blk * SK + (4 * s + 2 * g + 1)];
      acc = __builtin_amdgcn_wmma_f32_16x16x4_f32(
          false, A[s], false, b, (short)0, acc, false, false);
    }
    // store y2[i=m][j'=v+8g][k'=blk] back into addr(i,blk,j')
#pragma unroll
    for (int v = 0; v < 8; ++v)
      lds[m * SI + blk * SK + (v + 8 * g)] = acc[v];
  }

  // ---------------- Stage 3: contract i (LDS -> WMMA -> global) -------------
  // Tile blk = j'. B[i][n=k'] = y2[i=4s+2g+v][blk][k'=L&15]
  // at addr(i, k'=m, j'=blk).  Strides 2g*SI + m*SK are conflict-free mod 64.
#pragma unroll
  for (int blk = 0; blk < 16; ++blk) {
    v8f acc = {};
#pragma unroll
    for (int s = 0; s < 4; ++s) {
      v2f b;
      b.x = lds[(4 * s + 2 * g) * SI + m * SK + blk];
      b.y = lds[(4 * s + 2 * g + 1) * SI + m * SK + blk];
      acc = __builtin_amdgcn_wmma_f32_16x16x4_f32(
          false, A[s], false, b, (short)0, acc, false, false);
    }
    // D[m=i'][n=k'] = y[row][i'*256 + blk*16 + k']; lanes 0-15 / 16-31 each
    // write a contiguous 64B segment per vgpr -> coalesced streaming stores.
#pragma unroll
    for (int v = 0; v < 8; ++v)
      __builtin_nontemporal_store(acc[v], yo + (v + 8 * g) * 256 + blk * 16 + m);
  }
}

extern "C" void kernel_launch(void* const* d_in, const int* in_sizes, int n_in,
                              void* d_out, int out_size, void* d_ws, size_t ws_size,
                              hipStream_t stream) {
  const float* x = (const float*)d_in[0];
  float* y = (float*)d_out;
  const int rows = in_sizes[0] / 4096;                     // 8192
  const int blocks = (rows + WAVES_PER_BLOCK - 1) / WAVES_PER_BLOCK;
  wht4096_wmma_kernel<<<blocks, WAVES_PER_BLOCK * 32, 0, stream>>>(x, y, rows);
}